// GCN_42331197669874
// MI455X (gfx1250) — compile-verified
//
#include <hip/hip_runtime.h>

#define NCLS 64

typedef __attribute__((ext_vector_type(2))) float v2f;
typedef __attribute__((ext_vector_type(8))) float v8f;
typedef __attribute__((address_space(3))) char lds_char;

// ---------------------------------------------------------------------------
// Zero fill
// ---------------------------------------------------------------------------
__global__ void zero_kernel(float* __restrict__ p, long long n) {
  long long i = (long long)blockIdx.x * blockDim.x + threadIdx.x;
  if (i < n) p[i] = 0.0f;
}

// ---------------------------------------------------------------------------
// WMMA fp32 GEMM: C[MxN] = A[MxK] @ B[KxN]
// mode 0: plain store
// mode 1: C = 0.1*acc + 0.1*xw + 0.9*(agg + bias[n]); optional relu
// One wave per 16x16 output tile, 8 waves/block (16 rows x 128 cols slab).
// A tile staged into LDS with GLOBAL_LOAD_ASYNC_TO_LDS_B128 (ASYNCcnt path),
// K-loop uses two independent V_WMMA_F32_16X16X4_F32 accumulator chains.
// ---------------------------------------------------------------------------
__global__ void __launch_bounds__(256) gemm_wmma_f32(
    const float* __restrict__ A, const float* __restrict__ B,
    float* __restrict__ C, int M, int K, int kshift, int Nc, int mode,
    const float* __restrict__ xw, const float* __restrict__ agg,
    const float* __restrict__ bias, int relu) {
  __shared__ float As[16][132];  // pad to 132 -> conflict-free half-lane reads

  const int tid  = threadIdx.x;
  const int wave = tid >> 5;
  const int lane = tid & 31;
  const int half = lane >> 4;
  const int l16  = lane & 15;
  const int mBase = blockIdx.x * 16;

  if (mBase + 16 <= M) {
    // Async DMA of the 16xK fp32 A-tile straight into LDS (no VGPR staging).
    // GVS addressing: saddr = tile base (uniform), vaddr = 32-bit byte offset.
    const float* ablk = A + (size_t)mBase * K;
    const unsigned ldsBase = (unsigned)(unsigned long long)(lds_char*)&As[0][0];
    for (int idx4 = tid * 4; idx4 < 16 * K; idx4 += 256 * 4) {
      int r = idx4 >> kshift;
      int c = idx4 & (K - 1);
      unsigned ldsOff = ldsBase + (unsigned)(r * 132 + c) * 4u;
      unsigned gOff = (unsigned)idx4 * 4u;
      asm volatile("global_load_async_to_lds_b128 %0, %1, %2"
                   :
                   : "v"(ldsOff), "v"(gOff), "s"(ablk)
                   : "memory");
    }
    asm volatile("s_wait_asynccnt 0x0" ::: "memory");
  } else {
    for (int idx = tid; idx < 16 * K; idx += 256) {
      int r = idx >> kshift;
      int c = idx & (K - 1);
      int m = mBase + r;
      As[r][c] = (m < M) ? A[(size_t)m * K + c] : 0.0f;
    }
  }
  __syncthreads();

  const int nBase = wave * 16;
  if (nBase >= Nc) return;  // single barrier above; safe to exit

  // Two independent WMMA accumulation chains to break the RAW hazard chain.
  v8f acc0 = {};
  v8f acc1 = {};
  for (int k = 0; k < K; k += 8) {
    const int ka = k + 2 * half;
    v2f a0, b0;
    a0.x = As[l16][ka];
    a0.y = As[l16][ka + 1];
    const float* Bp0 = B + (size_t)ka * Nc + nBase + l16;
    b0.x = Bp0[0];
    b0.y = Bp0[Nc];
    acc0 = __builtin_amdgcn_wmma_f32_16x16x4_f32(false, a0, false, b0, (short)0,
                                                 acc0, false, false);
    const int kb = ka + 4;
    v2f a1, b1;
    a1.x = As[l16][kb];
    a1.y = As[l16][kb + 1];
    const float* Bp1 = Bp0 + (size_t)4 * Nc;
    b1.x = Bp1[0];
    b1.y = Bp1[Nc];
    acc1 = __builtin_amdgcn_wmma_f32_16x16x4_f32(false, a1, false, b1, (short)0,
                                                 acc1, false, false);
  }
  v8f acc = acc0 + acc1;

  const int n = nBase + l16;
#pragma unroll
  for (int j = 0; j < 8; ++j) {
    int m = mBase + j + 8 * half;
    if (m >= M) continue;
    size_t off = (size_t)m * Nc + n;
    float v = acc[j];
    if (mode == 1) {
      v = 0.1f * v + 0.1f * xw[off] + 0.9f * (agg[off] + bias[n]);
      if (relu) v = fmaxf(v, 0.0f);
    }
    C[off] = v;
  }
}

// ---------------------------------------------------------------------------
// Edge scatter: agg[dst[e]] += xw[src[e]]  (segment_sum). HBM/atomic bound.
// One wave covers one 128-float edge row (32 lanes x float4). cshift =
// log2(d/4) so the hot path is pure shift/mask (no integer division).
// ---------------------------------------------------------------------------
__global__ void scatter_add_kernel(const float* __restrict__ xw,
                                   const int* __restrict__ src,
                                   const int* __restrict__ dst,
                                   float* __restrict__ agg, int E, int d,
                                   int cshift) {
  long long gid = (long long)blockIdx.x * blockDim.x + threadIdx.x;
  long long total = (long long)E << cshift;
  if (gid >= total) return;
  int e = (int)(gid >> cshift);
  int c = (((int)gid & ((1 << cshift) - 1))) << 2;
  int s = src[e];
  int t = dst[e];
  // Prefetch a future gather row into cache (global_prefetch_b8 on gfx1250)
  if (e + 256 < E) {
    __builtin_prefetch(xw + (size_t)src[e + 256] * d + c, 0, 1);
  }
  const float4 v = *(const float4*)(xw + (size_t)s * d + c);
  float* o = agg + (size_t)t * d + c;
  atomicAdd(o + 0, v.x);
  atomicAdd(o + 1, v.y);
  atomicAdd(o + 2, v.z);
  atomicAdd(o + 3, v.w);
}

// ---------------------------------------------------------------------------
// Per-class sums + counts with LDS pre-aggregation (ds_add_f32) to avoid
// hammering 8K global addresses with 12.8M atomics. dshift = log2(d).
// ---------------------------------------------------------------------------
__global__ void __launch_bounds__(256) class_sums_kernel(
    const float* __restrict__ xw, const int* __restrict__ label,
    float* __restrict__ sums, float* __restrict__ counts, int Nn, int d,
    int dshift, int span) {
  __shared__ float sacc[NCLS * 128];
  __shared__ float scnt[NCLS];
  const int tid = threadIdx.x;
  for (int i = tid; i < NCLS * d; i += 256) sacc[i] = 0.0f;
  if (tid < NCLS) scnt[tid] = 0.0f;
  __syncthreads();

  const int n0 = blockIdx.x * span;
  const int n1 = min(n0 + span, Nn);
  const long long total = (long long)(n1 - n0) << dshift;
  for (long long idx = tid; idx < total; idx += 256) {
    int rel = (int)(idx >> dshift);
    int f = (int)idx & (d - 1);
    int n = n0 + rel;
    int lbl = label[n];
    atomicAdd(&sacc[(lbl << dshift) + f], xw[((size_t)n << dshift) + f]);
  }
  for (int n = n0 + tid; n < n1; n += 256) atomicAdd(&scnt[label[n]], 1.0f);
  __syncthreads();

  for (int i = tid; i < NCLS * d; i += 256) atomicAdd(&sums[i], sacc[i]);
  if (tid < NCLS) atomicAdd(&counts[tid], scnt[tid]);
}

// centers[c][f] = sums[c][f] / max(counts[c], 1)
__global__ void centers_kernel(const float* __restrict__ sums,
                               const float* __restrict__ counts,
                               float* __restrict__ centers, int d, int dshift) {
  int i = blockIdx.x * blockDim.x + threadIdx.x;
  if (i < NCLS * d) centers[i] = sums[i] / fmaxf(counts[i >> dshift], 1.0f);
}

// norms[i][j] = || centers[i] - centers[j] ||_2
__global__ void pair_norms_kernel(const float* __restrict__ centers,
                                  float* __restrict__ norms, int d) {
  int i = blockIdx.x;   // [0,64)
  int j = threadIdx.x;  // [0,64)
  float s = 0.0f;
  for (int f = 0; f < d; ++f) {
    float dl = centers[i * d + f] - centers[j * d + f];
    s += dl * dl;
  }
  norms[i * NCLS + j] = sqrtf(s);
}

// rc[i][f] = sum_j cm[i][j] * (centers[i][f]-centers[j][f]) / max(norm,!=0->1)
__global__ void rc_kernel(const float* __restrict__ centers,
                          const float* __restrict__ norms,
                          const float* __restrict__ cm, float* __restrict__ rc,
                          int d) {
  int i = blockIdx.x;
  int f = threadIdx.x;
  if (f >= d) return;
  float ci = centers[i * d + f];
  float acc = 0.0f;
  for (int j = 0; j < NCLS; ++j) {
    float nr = norms[i * NCLS + j];
    nr = (nr == 0.0f) ? 1.0f : nr;
    acc += cm[i * NCLS + j] * (ci - centers[j * d + f]) / nr;
  }
  rc[i * d + f] = acc;
}

// ---------------------------------------------------------------------------
// Host orchestration
// ---------------------------------------------------------------------------
extern "C" void kernel_launch(void* const* d_in, const int* in_sizes, int n_in,
                              void* d_out, int out_size, void* d_ws,
                              size_t ws_size, hipStream_t stream) {
  const float* x   = (const float*)d_in[0];
  const int*  edge = (const int*)d_in[1];
  const int* label = (const int*)d_in[2];
  const float* p   = (const float*)d_in[3];
  const float* cm  = (const float*)d_in[4];
  const float* W1  = (const float*)d_in[5];
  const float* b1  = (const float*)d_in[6];
  const float* W2  = (const float*)d_in[7];
  const float* b2  = (const float*)d_in[8];
  const float* W3  = (const float*)d_in[9];
  const float* b3  = (const float*)d_in[10];

  const int dIn = 128, dH = 128, dO = 64;
  const int Nn = in_sizes[0] / dIn;
  const int E  = in_sizes[1] / 2;
  const int* src = edge;
  const int* dst = edge + E;

  float* ws = (float*)d_ws;
  const size_t buf = (size_t)Nn * 128;
  float* B0 = ws;             // xw scratch
  float* B1 = B0 + buf;       // agg / h ping
  float* B2 = B1 + buf;       // agg / h pong
  float* sums    = B2 + buf;              // NCLS*128
  float* counts  = sums + NCLS * 128;     // NCLS
  float* centers = counts + NCLS;         // NCLS*128
  float* norms   = centers + NCLS * 128;  // NCLS*NCLS
  float* rcbuf   = norms + NCLS * NCLS;   // NCLS*128

  auto layer = [&](const float* hin, int dk, int dn, const float* W,
                   const float* bias, float* xwb, float* aggb, float* hout,
                   int relu) {
    const int kshiftA = __builtin_ctz(dk);
    const int dshift  = __builtin_ctz(dn);
    dim3 gGemm((unsigned)((Nn + 15) / 16), 1, 1);
    // 1: xw = hin @ W                         (WMMA fp32, async-LDS staged)
    gemm_wmma_f32<<<gGemm, 256, 0, stream>>>(hin, W, xwb, Nn, dk, kshiftA, dn,
                                             0, (const float*)nullptr,
                                             (const float*)nullptr,
                                             (const float*)nullptr, 0);
    // 2: zero agg + sums + counts
    long long nAgg = (long long)Nn * dn;
    zero_kernel<<<(unsigned)((nAgg + 255) / 256), 256, 0, stream>>>(aggb, nAgg);
    long long nSC = (long long)NCLS * 128 + NCLS;  // sums + counts contiguous
    zero_kernel<<<(unsigned)((nSC + 255) / 256), 256, 0, stream>>>(sums, nSC);
    // 3: agg[dst] += xw[src]                  (atomic scatter, HBM-bound)
    long long totalS = (long long)E << (dshift - 2);
    scatter_add_kernel<<<(unsigned)((totalS + 255) / 256), 256, 0, stream>>>(
        xwb, src, dst, aggb, E, dn, dshift - 2);
    // 4: per-class sums via LDS atomics
    const int span = 1024;
    class_sums_kernel<<<(unsigned)((Nn + span - 1) / span), 256, 0, stream>>>(
        xwb, label, sums, counts, Nn, dn, dshift, span);
    // 5: centers = sums / max(counts,1)
    int nC = NCLS * dn;
    centers_kernel<<<(unsigned)((nC + 255) / 256), 256, 0, stream>>>(
        sums, counts, centers, dn, dshift);
    // 6: pairwise center norms
    pair_norms_kernel<<<NCLS, NCLS, 0, stream>>>(centers, norms, dn);
    // 7: rc = cm-weighted normalized differences
    rc_kernel<<<NCLS, dn, 0, stream>>>(centers, norms, cm, rcbuf, dn);
    // 8: hout = 0.1*(p@rc) + 0.1*xw + 0.9*(agg+b)  (+relu), fused WMMA GEMM
    gemm_wmma_f32<<<gGemm, 256, 0, stream>>>(p, rcbuf, hout, Nn, NCLS,
                                             __builtin_ctz(NCLS), dn, 1, xwb,
                                             aggb, bias, relu);
  };

  // Layer 1: x -> B1 ; Layer 2: B1 -> B2 ; Layer 3: B2 -> d_out
  layer(x,  dIn, dH, W1, b1, B0, B1, B1, 1);
  layer(B1, dH,  dH, W2, b2, B0, B2, B2, 1);
  layer(B2, dH,  dO, W3, b3, B0, B1, (float*)d_out, 0);
}